// TGAA_MLP_42941083025520
// MI455X (gfx1250) — compile-verified
//
#include <hip/hip_runtime.h>

typedef float v2f __attribute__((ext_vector_type(2)));
typedef float v8f __attribute__((ext_vector_type(8)));

#define B_    512
#define D_    256
#define HID_  512
#define OUT_  128
#define CONC_ 1536   // 3*HID

// ---------------------------------------------------------------------------
// Phase 1a: inverse index map.  inv[idx[r]] = r   (inv pre-set to -1)
// ---------------------------------------------------------------------------
__global__ __launch_bounds__(256) void scatter_inv(const int* __restrict__ idx,
                                                   int nrows,
                                                   int* __restrict__ inv) {
  int r = blockIdx.x * 256 + threadIdx.x;
  if (r < nrows) inv[idx[r]] = r;
}

// ---------------------------------------------------------------------------
// Phase 1b: sum-pool.  Each block owns (batch b, 64-row chunk of positions),
// 256 threads = 256 feature columns.  Register accumulation, one f32 atomic
// per (element, chunk) at the end.  Reads the 235MB of x fully coalesced.
// ---------------------------------------------------------------------------
__global__ __launch_bounds__(256) void gather_sum(const float* __restrict__ x,
                                                  const int* __restrict__ inv,
                                                  int m, int chunks,
                                                  float* __restrict__ pooled) {
  const int b  = blockIdx.x / chunks;
  const int c  = blockIdx.x % chunks;
  const int d  = threadIdx.x;
  const int* invb = inv + b * m;

  int p0 = c * 64;
  int p1 = p0 + 64; if (p1 > m) p1 = m;

  float acc = 0.f;
  for (int p = p0; p < p1; ++p) {
    int r = invb[p];                       // uniform per block -> broadcast
    if (r >= 0) acc += x[(size_t)r * D_ + d];
  }
  unsafeAtomicAdd(&pooled[b * D_ + d], acc);  // global_atomic_add_f32
}

// ---------------------------------------------------------------------------
// Phase 2: h[:, dim*512 : (dim+1)*512] = relu(pooled_dim @ W1_dim + b1_dim)
// One wave -> 16(M) x 64(N) output strip, V_WMMA_F32_16X16X4_F32, K stepped by 4.
// A fragment reused across 4 WMMAs per K-step.
// ---------------------------------------------------------------------------
__global__ __launch_bounds__(256) void gemm1_relu(const float* __restrict__ pooledAll,
    const float* __restrict__ W1_0, const float* __restrict__ b1_0,
    const float* __restrict__ W1_1, const float* __restrict__ b1_1,
    const float* __restrict__ W1_2, const float* __restrict__ b1_2,
    float* __restrict__ h) {
  const int dim = blockIdx.y;
  const float* W    = (dim == 0) ? W1_0 : (dim == 1) ? W1_1 : W1_2;
  const float* bias = (dim == 0) ? b1_0 : (dim == 1) ? b1_1 : b1_2;
  const float* P    = pooledAll + (size_t)dim * B_ * D_;

  const int wave = threadIdx.x >> 5;
  const int lane = threadIdx.x & 31;
  const int job  = blockIdx.x * 8 + wave;   // 0..255
  const int mt   = job >> 3;                // 16-row tile  (0..31)
  const int ns   = job & 7;                 // 64-col strip (0..7)

  const int mrow = mt * 16 + (lane & 15);   // A row for this lane
  const int kk   = (lane >> 4) * 2;         // lane-half K offset within K4 group
  const int ncol = ns * 64 + (lane & 15);   // B column for this lane

  v8f a0 = {}, a1 = {}, a2 = {}, a3 = {};
  for (int k = 0; k < D_; k += 4) {
    v2f av = *(const v2f*)(P + (size_t)mrow * D_ + k + kk);
    const float* w0 = W + (size_t)(k + kk) * HID_ + ncol;
    const float* w1 = w0 + HID_;
    v2f bf0 = { w0[0],  w1[0]  };
    v2f bf1 = { w0[16], w1[16] };
    v2f bf2 = { w0[32], w1[32] };
    v2f bf3 = { w0[48], w1[48] };
    a0 = __builtin_amdgcn_wmma_f32_16x16x4_f32(false, av, false, bf0, (short)0, a0, false, false);
    a1 = __builtin_amdgcn_wmma_f32_16x16x4_f32(false, av, false, bf1, (short)0, a1, false, false);
    a2 = __builtin_amdgcn_wmma_f32_16x16x4_f32(false, av, false, bf2, (short)0, a2, false, false);
    a3 = __builtin_amdgcn_wmma_f32_16x16x4_f32(false, av, false, bf3, (short)0, a3, false, false);
  }

  const int nbase = ns * 64 + (lane & 15);
  const int rbase = mt * 16 + (lane >> 4) * 8;   // D: M = v + 8*(lane/16)
  const float bb0 = bias[nbase], bb1 = bias[nbase + 16],
              bb2 = bias[nbase + 32], bb3 = bias[nbase + 48];
#pragma unroll
  for (int v = 0; v < 8; ++v) {
    float* hr = h + (size_t)(rbase + v) * CONC_ + (size_t)dim * HID_ + nbase;
    hr[0]  = fmaxf(a0[v] + bb0, 0.f);
    hr[16] = fmaxf(a1[v] + bb1, 0.f);
    hr[32] = fmaxf(a2[v] + bb2, 0.f);
    hr[48] = fmaxf(a3[v] + bb3, 0.f);
  }
}

// ---------------------------------------------------------------------------
// Phase 3: out = h (512x1536) @ W2 (1536x128) + b2
// ---------------------------------------------------------------------------
__global__ __launch_bounds__(256) void gemm2(const float* __restrict__ h,
                                             const float* __restrict__ W2,
                                             const float* __restrict__ b2,
                                             float* __restrict__ out) {
  const int wave = threadIdx.x >> 5;
  const int lane = threadIdx.x & 31;
  const int job  = blockIdx.x * 8 + wave;   // 0..63
  const int mt   = job >> 1;                // 0..31
  const int ns   = job & 1;                 // 0..1

  const int mrow = mt * 16 + (lane & 15);
  const int kk   = (lane >> 4) * 2;
  const int ncol = ns * 64 + (lane & 15);

  v8f a0 = {}, a1 = {}, a2 = {}, a3 = {};
  for (int k = 0; k < CONC_; k += 4) {
    v2f av = *(const v2f*)(h + (size_t)mrow * CONC_ + k + kk);
    const float* w0 = W2 + (size_t)(k + kk) * OUT_ + ncol;
    const float* w1 = w0 + OUT_;
    v2f bf0 = { w0[0],  w1[0]  };
    v2f bf1 = { w0[16], w1[16] };
    v2f bf2 = { w0[32], w1[32] };
    v2f bf3 = { w0[48], w1[48] };
    a0 = __builtin_amdgcn_wmma_f32_16x16x4_f32(false, av, false, bf0, (short)0, a0, false, false);
    a1 = __builtin_amdgcn_wmma_f32_16x16x4_f32(false, av, false, bf1, (short)0, a1, false, false);
    a2 = __builtin_amdgcn_wmma_f32_16x16x4_f32(false, av, false, bf2, (short)0, a2, false, false);
    a3 = __builtin_amdgcn_wmma_f32_16x16x4_f32(false, av, false, bf3, (short)0, a3, false, false);
  }

  const int nbase = ns * 64 + (lane & 15);
  const int rbase = mt * 16 + (lane >> 4) * 8;
  const float bb0 = b2[nbase], bb1 = b2[nbase + 16],
              bb2 = b2[nbase + 32], bb3 = b2[nbase + 48];
#pragma unroll
  for (int v = 0; v < 8; ++v) {
    float* orow = out + (size_t)(rbase + v) * OUT_ + nbase;
    orow[0]  = a0[v] + bb0;
    orow[16] = a1[v] + bb1;
    orow[32] = a2[v] + bb2;
    orow[48] = a3[v] + bb3;
  }
}

// ---------------------------------------------------------------------------
extern "C" void kernel_launch(void* const* d_in, const int* in_sizes, int n_in,
                              void* d_out, int out_size, void* d_ws, size_t ws_size,
                              hipStream_t stream) {
  const float* x0   = (const float*)d_in[0];
  const int*   idx0 = (const int*)  d_in[1];
  const float* x1   = (const float*)d_in[2];
  const int*   idx1 = (const int*)  d_in[3];
  const float* x2   = (const float*)d_in[4];
  const int*   idx2 = (const int*)  d_in[5];
  const float* W1_0 = (const float*)d_in[6];
  const float* b1_0 = (const float*)d_in[7];
  const float* W1_1 = (const float*)d_in[8];
  const float* b1_1 = (const float*)d_in[9];
  const float* W1_2 = (const float*)d_in[10];
  const float* b1_2 = (const float*)d_in[11];
  const float* W2   = (const float*)d_in[12];
  const float* b2   = (const float*)d_in[13];
  float* out = (float*)d_out;

  // workspace layout (floats): pooled[3*512*256] | h[512*1536] | inv (ints)
  float* pooled = (float*)d_ws;
  float* h      = pooled + 3 * B_ * D_;
  int*   inv    = (int*)(h + (size_t)B_ * CONC_);
  const int n0 = B_ * 128, n1 = B_ * 256, n2 = B_ * 64;   // slots per dim
  int* inv0 = inv;
  int* inv1 = inv0 + n0;
  int* inv2 = inv1 + n1;

  hipMemsetAsync(pooled, 0,    (size_t)3 * B_ * D_ * sizeof(float), stream);
  hipMemsetAsync(inv,    0xFF, (size_t)(n0 + n1 + n2) * sizeof(int), stream);

  scatter_inv<<<(n0 + 255) / 256, 256, 0, stream>>>(idx0, n0, inv0);
  scatter_inv<<<(n1 + 255) / 256, 256, 0, stream>>>(idx1, n1, inv1);
  scatter_inv<<<(n2 + 255) / 256, 256, 0, stream>>>(idx2, n2, inv2);

  gather_sum<<<B_ * 2, 256, 0, stream>>>(x0, inv0, 128, 2, pooled);
  gather_sum<<<B_ * 4, 256, 0, stream>>>(x1, inv1, 256, 4, pooled + B_ * D_);
  gather_sum<<<B_ * 1, 256, 0, stream>>>(x2, inv2,  64, 1, pooled + 2 * B_ * D_);

  dim3 g1(32, 3);  // 32 blocks x 8 waves = 256 wave-jobs per dim
  gemm1_relu<<<g1, 256, 0, stream>>>(pooled, W1_0, b1_0, W1_1, b1_1, W1_2, b1_2, h);

  gemm2<<<8, 256, 0, stream>>>(h, W2, b2, out);
}